// MedianConv_41137196761671
// MI455X (gfx1250) — compile-verified
//
#include <hip/hip_runtime.h>

// MedianConv for MI455X (gfx1250, wave32).
//
// Phase 1: h = x @ W via V_WMMA_F32_16X16X4_F32 (exact f32 accumulate).
//   - one wave computes a 16(M) x 64(N) strip of h; K=128 in steps of 4
//   - W (128x64, 32KB) staged into LDS transposed with row stride 132 floats
//     -> bank-conflict-free ds_load_b64 B-fragment reads
// Phase 2: per-node median of 33 gathered h-rows.
//   - one wave per node; 32 neighbor indices in one coalesced 128B load
//   - each lane owns channels (2*lane, 2*lane+1); per neighbor the wave reads
//     the full 256B h row coalesced (float2 per lane)
//   - exact median-of-33 = 17th smallest, kept via unrolled sorted-17
//     insertion chain of v_min/v_max (branchless, deterministic)

typedef __attribute__((ext_vector_type(2))) float v2f;
typedef __attribute__((ext_vector_type(8))) float v8f;

#define IN_DIM 128
#define OUT_DIM 64
#define KNBR 32
#define LDS_STRIDE 132   // 64*132 floats; (col*132)%64 distinct per lane group

__global__ __launch_bounds__(256) void gemm_wmma_f32_kernel(
    const float* __restrict__ x, const float* __restrict__ W,
    float* __restrict__ h, int N) {
  __shared__ float Wt[OUT_DIM * LDS_STRIDE];  // W transposed: Wt[c*132 + k]
  const int tid = threadIdx.x;

  // cooperative, coalesced global read of W; transpose into padded LDS
  for (int idx = tid; idx < IN_DIM * OUT_DIM; idx += 256) {
    int kk = idx >> 6;   // idx / 64  (K index)
    int c  = idx & 63;   // channel
    Wt[c * LDS_STRIDE + kk] = W[idx];
  }
  __syncthreads();

  const int lane = tid & 31;
  const int half = lane >> 4;   // 0: lanes 0-15, 1: lanes 16-31
  const int l    = lane & 15;
  const int rowTile = blockIdx.x * 8 + (tid >> 5);
  const int row0 = rowTile * 16;
  if (row0 >= N) return;

  if (row0 + 16 <= N) {
    v8f acc0 = {}, acc1 = {}, acc2 = {}, acc3 = {};
    const float* xrow = x + (size_t)(row0 + l) * IN_DIM;
#pragma unroll 4
    for (int k = 0; k < IN_DIM; k += 4) {
      const int kk = k + 2 * half;
      // A fragment: 16x4 f32 -> lane(l,half) holds x[row0+l][kk], x[row0+l][kk+1]
      v2f a = *(const v2f*)(xrow + kk);
      // B fragments: 4x16 f32 per col tile, K striping mirrors A
      v2f b0 = *(const v2f*)(&Wt[(0 * 16 + l) * LDS_STRIDE + kk]);
      v2f b1 = *(const v2f*)(&Wt[(1 * 16 + l) * LDS_STRIDE + kk]);
      v2f b2 = *(const v2f*)(&Wt[(2 * 16 + l) * LDS_STRIDE + kk]);
      v2f b3 = *(const v2f*)(&Wt[(3 * 16 + l) * LDS_STRIDE + kk]);
      acc0 = __builtin_amdgcn_wmma_f32_16x16x4_f32(false, a, false, b0, (short)0, acc0, false, false);
      acc1 = __builtin_amdgcn_wmma_f32_16x16x4_f32(false, a, false, b1, (short)0, acc1, false, false);
      acc2 = __builtin_amdgcn_wmma_f32_16x16x4_f32(false, a, false, b2, (short)0, acc2, false, false);
      acc3 = __builtin_amdgcn_wmma_f32_16x16x4_f32(false, a, false, b3, (short)0, acc3, false, false);
    }
    // C/D layout: VGPR v -> row = row0 + v + 8*half, col = l (+16 per tile)
#pragma unroll
    for (int v = 0; v < 8; ++v) {
      float* hr = h + (size_t)(row0 + v + 8 * half) * OUT_DIM;
      hr[ 0 + l] = acc0[v];
      hr[16 + l] = acc1[v];
      hr[32 + l] = acc2[v];
      hr[48 + l] = acc3[v];
    }
  } else {
    // remainder tile (not hit for N=100000, kept for robustness)
    const int r = row0 + l;
    if (half == 0 && r < N) {
      for (int c = 0; c < OUT_DIM; ++c) {
        float s = 0.f;
        for (int k = 0; k < IN_DIM; ++k)
          s = fmaf(x[(size_t)r * IN_DIM + k], Wt[c * LDS_STRIDE + k], s);
        h[(size_t)r * OUT_DIM + c] = s;
      }
    }
  }
}

// keep the 17 smallest values seen so far, sorted ascending, in registers
__device__ __forceinline__ void insert17(float (&m)[17], float v) {
  float c = v;
#pragma unroll
  for (int j = 0; j < 17; ++j) {
    float lo = fminf(m[j], c);
    float hi = fmaxf(m[j], c);
    m[j] = lo;
    c = hi;
  }
}

__global__ __launch_bounds__(256) void median_gather_kernel(
    const float* __restrict__ h, const int* __restrict__ row,
    const float* __restrict__ bias, float* __restrict__ out, int N) {
  const int wave = blockIdx.x * (blockDim.x >> 5) + (threadIdx.x >> 5);
  const int lane = threadIdx.x & 31;
  if (wave >= N) return;
  const int i = wave;

  // one coalesced 128B load: lane holds one of the 32 neighbor indices
  const int myIdx = row[(size_t)i * KNBR + lane];

  float m0[17], m1[17];
#pragma unroll
  for (int j = 0; j < 17; ++j) { m0[j] = __builtin_inff(); m1[j] = __builtin_inff(); }

  for (int j = 0; j < KNBR; ++j) {
    const int nbr = __shfl(myIdx, j, 32);            // ds_bpermute broadcast
    const float2 v = ((const float2*)(h + (size_t)nbr * OUT_DIM))[lane];
    insert17(m0, v.x);
    insert17(m1, v.y);
  }
  {  // self loop (33rd value)
    const float2 v = ((const float2*)(h + (size_t)i * OUT_DIM))[lane];
    insert17(m0, v.x);
    insert17(m1, v.y);
  }

  const float2 bb = ((const float2*)bias)[lane];
  float2 o;
  o.x = m0[16] + bb.x;   // 17th smallest of 33 = exact median
  o.y = m1[16] + bb.y;
  ((float2*)(out + (size_t)i * OUT_DIM))[lane] = o;
}

extern "C" void kernel_launch(void* const* d_in, const int* in_sizes, int n_in,
                              void* d_out, int out_size, void* d_ws, size_t ws_size,
                              hipStream_t stream) {
  const float* x    = (const float*)d_in[0];  // [N, 128]
  const int*   edge = (const int*)d_in[1];    // [2, E]; first E entries = row
  const float* W    = (const float*)d_in[2];  // [128, 64]
  const float* b    = (const float*)d_in[3];  // [64]
  float* out = (float*)d_out;                 // [N, 64]
  float* h   = (float*)d_ws;                  // [N, 64] scratch

  const int N = in_sizes[0] / IN_DIM;         // 100000

  const int rowTiles = (N + 15) / 16;
  const int gemmBlocks = (rowTiles + 7) / 8;  // 8 waves/block, 1 tile/wave
  gemm_wmma_f32_kernel<<<gemmBlocks, 256, 0, stream>>>(x, W, h, N);

  const int medBlocks = (N + 7) / 8;          // 8 waves/block, 1 node/wave
  median_gather_kernel<<<medBlocks, 256, 0, stream>>>(h, edge, b, out, N);
}